// VRTRPostProcess_55937654063234
// MI455X (gfx1250) — compile-verified
//
#include <hip/hip_runtime.h>
#include <stdint.h>

// ---------------------------------------------------------------------------
// Problem constants (from reference):  B=32, Q=300, C=81, P=300, A=29
// ---------------------------------------------------------------------------
#define B_ 32
#define Q_ 300
#define C_ 81
#define P_ 300
#define A_ 29

// Flat output layout (all float32, concatenated in reference return order):
//   scores  (B,Q)           -> [0,            9600)
//   labels  (B,Q)           -> [9600,         19200)   (indices stored as float)
//   boxes   (B,Q,4)         -> [19200,        57600)
//   score   (B,A,Q,Q+1)     -> [57600,        83856000)
//   h_mask  (B,Q)           -> [83856000,     83865600)
//   o_mask  (B,Q+1)         -> [83865600,     83875232)
#define OFF_SCORES   0
#define OFF_LABELS   (B_ * Q_)
#define OFF_BOXES    (2 * B_ * Q_)
#define OFF_SCORE    (OFF_BOXES + B_ * Q_ * 4)            // 57600
#define SCORE_ELEMS  ((size_t)B_ * A_ * Q_ * (Q_ + 1))    // 83,798,400
#define OFF_HMASK    (OFF_SCORE + SCORE_ELEMS)
#define OFF_OMASK    (OFF_HMASK + (size_t)B_ * Q_)

// ---------------------------------------------------------------------------
// TDM zero-fill of the big score tensor.
// One (b,a) plane = 300*301 = 90300 floats = 361,200 B per block.
// Per block: zero 15050 floats (60,200 B) of LDS once, then wave 0 issues
// 6 TENSOR_STORE_FROM_LDS DMAs (1-D tile, tile_dim0 = 15050, data_size = 4B).
// 928 blocks * 6 * 15050 = 83,798,400 floats: exact coverage, no OOB needed.
// ---------------------------------------------------------------------------
#define CHUNK_FLOATS     15050
#define CHUNK_BYTES      (CHUNK_FLOATS * 4)   // 60,200
#define CHUNKS_PER_BLOCK 6
#define ZFILL_BLOCKS     (B_ * A_)            // 928

typedef __attribute__((ext_vector_type(4))) unsigned int u32x4;
typedef __attribute__((ext_vector_type(8))) int          i32x8;
typedef __attribute__((ext_vector_type(4))) int          i32x4;

__global__ void __launch_bounds__(256)
vrtr_zfill_tdm_kernel(float* __restrict__ score_base) {
    __shared__ float zbuf[CHUNK_FLOATS];   // sole LDS object -> LDS offset 0

    for (int i = threadIdx.x; i < CHUNK_FLOATS; i += 256)
        zbuf[i] = 0.0f;

    // The TDM intrinsic receives lds_addr as a literal (0), so zbuf's address
    // never escapes and the compiler would dead-store-eliminate the zeroing
    // loop (observed in round-1 asm: ds=0). Escape the pointer into an asm
    // with a memory clobber so the LDS stores are considered live.
    asm volatile("" : : "r"(&zbuf[0]) : "memory");

    __syncthreads();   // LDS zeros visible before the DMA engine reads them

    // TDM issues once per wave regardless of EXEC; only wave 0 may reach it.
    if (threadIdx.x < 32) {
        uint64_t base = (uint64_t)(uintptr_t)score_base +
                        (uint64_t)blockIdx.x * ((uint64_t)CHUNKS_PER_BLOCK * CHUNK_BYTES);

        // D# group1 (256b): workgroup_mask=0, data_size=2 (4B), no barrier/
        // iterate/pad.  tensor_dim0 = tile_dim0 = 15050, tensor_dim0_stride =
        // 15050, tensor_dim1 = 1, tile_dim1/2 = 0 (unused -> 1-D tile).
        i32x8 g1 = { (int)0x00020000u,                    // bits17:16 data_size=2
                     (int)((unsigned)CHUNK_FLOATS << 16), // tensor_dim0[15:0]
                     (int)(1u << 16),                     // tensor_dim1[15:0] = 1
                     (int)((unsigned)CHUNK_FLOATS << 16), // tile_dim0
                     0,                                   // tile_dim1=0, tile_dim2=0
                     (int)CHUNK_FLOATS,                   // tensor_dim0_stride[31:0]
                     0, 0 };
        i32x4 gz = { 0, 0, 0, 0 };                        // dims >=2 unused
#if __clang_major__ >= 23
        i32x8 gz8 = { 0, 0, 0, 0, 0, 0, 0, 0 };
#endif

        #pragma unroll
        for (int j = 0; j < CHUNKS_PER_BLOCK; ++j) {
            uint64_t ga = base + (uint64_t)j * CHUNK_BYTES;
            u32x4 g0;
            g0.x = 1u;                                  // count=1, user descriptor
            g0.y = 0u;                                  // lds_addr = 0 (zbuf)
            g0.z = (unsigned)(ga & 0xFFFFFFFFu);        // global_addr[31:0]
            g0.w = (unsigned)((ga >> 32) & 0x1FFFFFFu)  // global_addr[56:32]
                 | (2u << 30);                          // type = 2 ("image")
#if __clang_major__ >= 23
            __builtin_amdgcn_tensor_store_from_lds(g0, g1, gz, gz, gz8, 0);
#else
            __builtin_amdgcn_tensor_store_from_lds(g0, g1, gz, gz, 0);
#endif
        }
        __builtin_amdgcn_s_wait_tensorcnt(0);  // drain TENSORcnt before exit
    }
}

// ---------------------------------------------------------------------------
// Per-query head: softmax max/argmax over 81 classes, scaled boxes, masks.
// One thread per (b,q); first B threads also write o_mask's appended column.
// ---------------------------------------------------------------------------
__global__ void __launch_bounds__(256)
vrtr_head_kernel(const float* __restrict__ logits,   // (B,Q,81)
                 const float* __restrict__ boxes_in, // (B,Q,4) cxcywh
                 const float* __restrict__ tsizes,   // (B,2)   (h,w)
                 float* __restrict__ out) {
    int i = blockIdx.x * blockDim.x + threadIdx.x;

    if (i < B_)  // o_mask last column = True
        out[OFF_OMASK + (size_t)i * (Q_ + 1) + Q_] = 1.0f;

    if (i >= B_ * Q_) return;
    const int b = i / Q_;
    const int q = i - b * Q_;

    const float* l = logits + (size_t)i * C_;

    // numerically stable softmax over all 81 classes
    float m = l[0];
    #pragma unroll 1
    for (int c = 1; c < C_; ++c) m = fmaxf(m, l[c]);

    float z = 0.0f;
    float m80 = l[0];
    int   arg = 0;
    #pragma unroll 1
    for (int c = 0; c < C_; ++c) {
        const float v = l[c];
        z += expf(v - m);
        if (c > 0 && c < C_ - 1 && v > m80) { m80 = v; arg = c; }
    }
    const float score = expf(m80 - m) / z;   // max prob over first 80 classes

    out[OFF_SCORES + i] = score;
    out[OFF_LABELS + i] = (float)arg;

    // boxes: cxcywh -> xyxy, scaled by (w,h,w,h); target_sizes = (h,w)
    const float img_h = tsizes[b * 2 + 0];
    const float img_w = tsizes[b * 2 + 1];
    const float cx = boxes_in[(size_t)i * 4 + 0];
    const float cy = boxes_in[(size_t)i * 4 + 1];
    const float w  = boxes_in[(size_t)i * 4 + 2];
    const float h  = boxes_in[(size_t)i * 4 + 3];
    float* bo = out + OFF_BOXES + (size_t)i * 4;
    bo[0] = (cx - 0.5f * w) * img_w;
    bo[1] = (cy - 0.5f * h) * img_h;
    bo[2] = (cx + 0.5f * w) * img_w;
    bo[3] = (cy + 0.5f * h) * img_h;

    const bool pos = score > 0.0f;                       // threshold = 0.0
    out[OFF_HMASK + i] = (arg == 1 && pos) ? 1.0f : 0.0f;
    out[OFF_OMASK + (size_t)b * (Q_ + 1) + q] = pos ? 1.0f : 0.0f;
}

// ---------------------------------------------------------------------------
// Scatter sigmoid(pred_actions) into the transposed (B,A,Q,Q+1) tensor.
// One thread per (b,p); 29 strided stores each. Runs after the TDM zero-fill.
// ---------------------------------------------------------------------------
__global__ void __launch_bounds__(256)
vrtr_scatter_kernel(const float* __restrict__ acts,   // (B,P,A)
                    const int*   __restrict__ pairs,  // (B,P,2)
                    float* __restrict__ out) {
    int i = blockIdx.x * blockDim.x + threadIdx.x;
    if (i >= B_ * P_) return;
    const int b = i / P_;

    int h = pairs[2 * i + 0];
    int o = pairs[2 * i + 1];
    if (h == o) o = Q_;                    // diagonal pairs -> appended column

    const float* a = acts + (size_t)i * A_;
    float* base = out + OFF_SCORE + (size_t)b * A_ * Q_ * (Q_ + 1);
    const size_t ho = (size_t)h * (Q_ + 1) + (size_t)o;

    #pragma unroll 1
    for (int k = 0; k < A_; ++k) {
        const float s = 1.0f / (1.0f + expf(-a[k]));
        base[(size_t)k * Q_ * (Q_ + 1) + ho] = s;
    }
}

// ---------------------------------------------------------------------------
// Launch: fill (TDM) -> head -> scatter, serialized on `stream`.
// ---------------------------------------------------------------------------
extern "C" void kernel_launch(void* const* d_in, const int* in_sizes, int n_in,
                              void* d_out, int out_size, void* d_ws, size_t ws_size,
                              hipStream_t stream) {
    const float* pred_logits    = (const float*)d_in[0];  // (B,Q,81)
    const float* pred_boxes     = (const float*)d_in[1];  // (B,Q,4)
    const float* pred_actions   = (const float*)d_in[2];  // (B,P,A)
    const int*   pred_rel_pairs = (const int*)  d_in[3];  // (B,P,2)
    const float* target_sizes   = (const float*)d_in[4];  // (B,2)
    float* out = (float*)d_out;

    // 1) Zero the 335 MB score tensor via the Tensor Data Mover.
    vrtr_zfill_tdm_kernel<<<ZFILL_BLOCKS, 256, 0, stream>>>(out + OFF_SCORE);

    // 2) Per-query outputs (independent of the fill; ordering is harmless).
    vrtr_head_kernel<<<(B_ * Q_ + 255) / 256, 256, 0, stream>>>(
        pred_logits, pred_boxes, target_sizes, out);

    // 3) Scatter the 278K sigmoid values (stream order guarantees the fill
    //    finished first).
    vrtr_scatter_kernel<<<(B_ * P_ + 255) / 256, 256, 0, stream>>>(
        pred_actions, pred_rel_pairs, out);
}